// NCCLayer_40261023433250
// MI455X (gfx1250) — compile-verified
//
#include <hip/hip_runtime.h>
#include <cstdint>
#include <cstddef>

// Volume: (N=2, D=160, H=192, W=160, C=1), f32. 9x9x9 box sums, SAME (zero) pad.
#define NB 2
#define DD 160
#define HHH 192
#define WWW 160

// Tile: each block -> (TH x TW) output tile, streaming a TD-deep D segment.
#define TD 40
#define TH 8
#define TW 32
#define NPLANES (TD + 8)          // 48 raw d-planes per block (4-halo each side)
#define RH (TH + 8)               // 16 raw rows  (h-halo)
#define RW (TW + 8)               // 40 raw cols  (w-halo), 16B-aligned segments
#define NBLOCKS (NB * (DD / TD) * (HHH / TH) * (WWW / TW))   // 2*4*24*5 = 960
#define WIN_F 729.0f

typedef __attribute__((ext_vector_type(2))) float v2f;
typedef __attribute__((ext_vector_type(8))) float v8f;

// Generic pointers into LDS are {shared_aperture_hi32 | lds_offset_lo32};
// low 32 bits are the wave-LDS-base-relative byte offset the async op needs.
__device__ __forceinline__ unsigned lds_byte_off(const void* p) {
  return (unsigned)(unsigned long long)p;
}

// CDNA5 async copy global -> LDS (tracked by ASYNCcnt). GV mode, 64-bit vaddr.
__device__ __forceinline__ void async_load_b128(unsigned ldsOff, const float* gp) {
  asm volatile("global_load_async_to_lds_b128 %0, %1, off"
               :: "v"(ldsOff), "v"((unsigned long long)gp)
               : "memory");
}

__global__ __launch_bounds__(256) void ncc_fused_kernel(
    const float* __restrict__ I, const float* __restrict__ J,
    float* __restrict__ partials)
{
  // Double-buffered raw planes + per-plane W-summed channels. ~20 KB of 320 KB/WGP.
  __shared__ __align__(16) float sIp[2][RH][RW];
  __shared__ __align__(16) float sJp[2][RH][RW];
  __shared__ float sWp[5][RH][TW];
  __shared__ __align__(16) float sDummy[4];   // sink for OOB async lanes
  __shared__ float sRed[8];                   // one partial per wave32

  const int t = threadIdx.x;
  int b = blockIdx.x;
  const int wt = b % (WWW / TW); b /= (WWW / TW);
  const int ht = b % (HHH / TH); b /= (HHH / TH);
  const int dt = b % (DD / TD);  b /= (DD / TD);
  const int nn = b;
  const int d0 = dt * TD, h0 = ht * TH, w0 = wt * TW;

  // Raw loader geometry: threads 0..159 move one 16B segment of I and of J.
  // (Waves 0..4 issue exactly 2 async ops/plane each; waves 5..7 issue none,
  //  so per-wave ASYNCcnt bookkeeping stays uniform and deterministic.)
  const int rrow = t / 10;
  const int seg  = t - rrow * 10;
  const int gh   = h0 - 4 + rrow;
  const int gw   = w0 - 4 + seg * 4;   // 4-float aligned; OOB is all-or-nothing

  auto issue_plane = [&](int p) {
    if (t < 160) {
      const int gd = d0 - 4 + p;
      const bool inb = ((unsigned)gd < (unsigned)DD) &&
                       ((unsigned)gh < (unsigned)HHH) &&
                       ((unsigned)gw < (unsigned)WWW);
      const int cd = inb ? gd : 0, ch = inb ? gh : 0, cw = inb ? gw : 0;
      const size_t gidx = ((size_t)((nn * DD + cd) * HHH + ch)) * WWW + cw;
      const int buf = p & 1;
      const unsigned offI = inb ? lds_byte_off(&sIp[buf][rrow][seg * 4])
                                : lds_byte_off(sDummy);
      const unsigned offJ = inb ? lds_byte_off(&sJp[buf][rrow][seg * 4])
                                : lds_byte_off(sDummy);
      async_load_b128(offI, I + gidx);
      async_load_b128(offJ, J + gidx);
      if (!inb) {  // zero-pad halo (disjoint addresses from the async writes)
        float4 z; z.x = z.y = z.z = z.w = 0.f;
        *reinterpret_cast<float4*>(&sIp[buf][rrow][seg * 4]) = z;
        *reinterpret_cast<float4*>(&sJp[buf][rrow][seg * 4]) = z;
      }
    }
  };

  // D-direction ring buffer (registers): last 9 H&W-summed plane values, 5 ch.
  float ring[5][9];
  float run[5];
  float acc = 0.f;
#pragma unroll
  for (int c = 0; c < 5; ++c) {
    run[c] = 0.f;
#pragma unroll
    for (int k = 0; k < 9; ++k) ring[c][k] = 0.f;
  }

  issue_plane(0);  // prime the pipeline

  for (int p = 0; p < NPLANES; ++p) {
    if (p + 1 < NPLANES) {
      issue_plane(p + 1);                               // prefetch next plane
      asm volatile("s_wait_asynccnt 2" ::: "memory");   // plane p resident
    } else {
      asm volatile("s_wait_asynccnt 0" ::: "memory");
    }
    __syncthreads();

    const int buf = p & 1;

    // ---- Stage W: products + 9-tap w-sum. 2 adjacent outputs/thread (tap reuse).
    {
      const int hh = t >> 4;            // 0..15
      const int x0 = (t & 15) << 1;     // 0,2,..,30
      float a0=0,a1=0,a2=0,a3=0,a4=0, c0=0,c1=0,c2=0,c3=0,c4=0;
#pragma unroll
      for (int tw = 0; tw < 10; ++tw) {
        const float iv = sIp[buf][hh][x0 + tw];
        const float jv = sJp[buf][hh][x0 + tw];
        const float pii = iv * iv, pjj = jv * jv, pij = iv * jv;
        if (tw < 9) { a0+=iv; a1+=jv; a2+=pii; a3+=pjj; a4+=pij; }
        if (tw > 0) { c0+=iv; c1+=jv; c2+=pii; c3+=pjj; c4+=pij; }
      }
      sWp[0][hh][x0] = a0; sWp[1][hh][x0] = a1; sWp[2][hh][x0] = a2;
      sWp[3][hh][x0] = a3; sWp[4][hh][x0] = a4;
      sWp[0][hh][x0+1] = c0; sWp[1][hh][x0+1] = c1; sWp[2][hh][x0+1] = c2;
      sWp[3][hh][x0+1] = c3; sWp[4][hh][x0+1] = c4;
    }
    __syncthreads();

    // ---- Stage H (9-tap h-sum) + D ring update (registers), cc emit.
    {
      const int oh = t >> 5;   // 0..7
      const int ox = t & 31;   // 0..31
      float v[5] = {0.f, 0.f, 0.f, 0.f, 0.f};
#pragma unroll
      for (int th2 = 0; th2 < 9; ++th2) {
#pragma unroll
        for (int c = 0; c < 5; ++c) v[c] += sWp[c][oh + th2][ox];
      }
#pragma unroll
      for (int c = 0; c < 5; ++c) run[c] += v[c] - ring[c][0];  // add new, drop d-9

      if (p >= 8) {  // window [p-8, p] complete -> output voxel d = d0 + p - 8
        const float inv = 1.0f / WIN_F;
        const float uI = run[0] * inv, uJ = run[1] * inv;
        const float cross = run[4] - uJ * run[0] - uI * run[1] + uI * uJ * WIN_F;
        const float Iv = run[2] - 2.0f * uI * run[0] + uI * uI * WIN_F;
        const float Jv = run[3] - 2.0f * uJ * run[1] + uJ * uJ * WIN_F;
        acc += (cross * cross) / (Iv * Jv + 1e-5f);
      }
#pragma unroll
      for (int c = 0; c < 5; ++c) {   // shift ring, append plane p
#pragma unroll
        for (int k = 0; k < 8; ++k) ring[c][k] = ring[c][k + 1];
        ring[c][8] = v[c];
      }
    }
    __syncthreads();  // stageH reads of sWp done before next plane's stageW writes
  }

  // ---- Block reduction. Per-wave sum via V_WMMA_F32_16X16X4_F32 with B = ones
  // (all-ones B is layout-invariant). A layout (ISA 16x4 f32): lane l<16 holds
  // A[l][0] (v0) / A[l][1] (v1); lane l>=16 holds A[l-16][2] / A[l-16][3].
  // With A.x = acc, A.y = 0:  D[m][n] = acc(m) + acc(m+16)  (same for all n).
  // Lane-half row sums + one xor-16 shuffle give the full wave32 sum, exactly
  // (f32 RNE, fixed order -> deterministic). EXEC is all-ones here as required.
  {
    const int lane = t & 31;
    v2f A; A.x = acc;  A.y = 0.0f;
    v2f B; B.x = 1.0f; B.y = 1.0f;
    v8f C = {0.f, 0.f, 0.f, 0.f, 0.f, 0.f, 0.f, 0.f};
    v8f Dm = __builtin_amdgcn_wmma_f32_16x16x4_f32(
        /*neg_a=*/false, A, /*neg_b=*/false, B,
        /*c_mod=*/(short)0, C, /*reuse_a=*/false, /*reuse_b=*/false);
    float half8 = ((Dm[0] + Dm[1]) + (Dm[2] + Dm[3])) +
                  ((Dm[4] + Dm[5]) + (Dm[6] + Dm[7]));   // 8 M-rows of this half
    float wtot = half8 + __shfl_xor(half8, 16, 32);       // + other half's rows
    if (lane == 0) sRed[t >> 5] = wtot;
  }
  __syncthreads();
  if (t == 0) {
    float s = 0.f;
#pragma unroll
    for (int w = 0; w < 8; ++w) s += sRed[w];
    partials[blockIdx.x] = s;
  }
}

__global__ __launch_bounds__(256) void ncc_finalize_kernel(
    const float* __restrict__ partials, int n, float* __restrict__ out)
{
  __shared__ double red[256];
  const int t = threadIdx.x;
  double s = 0.0;
  for (int i = t; i < n; i += 256) s += (double)partials[i];
  red[t] = s;
  __syncthreads();
  for (int k = 128; k > 0; k >>= 1) {
    if (t < k) red[t] += red[t + k];
    __syncthreads();
  }
  if (t == 0) {
    const double total = (double)NB * DD * HHH * WWW;
    out[0] = (float)(1.0 - red[0] / total);
  }
}

extern "C" void kernel_launch(void* const* d_in, const int* in_sizes, int n_in,
                              void* d_out, int out_size, void* d_ws, size_t ws_size,
                              hipStream_t stream) {
  const float* I = (const float*)d_in[0];
  const float* J = (const float*)d_in[1];
  float* out = (float*)d_out;
  float* partials = (float*)d_ws;   // NBLOCKS floats (3.8 KB) of scratch

  ncc_fused_kernel<<<NBLOCKS, 256, 0, stream>>>(I, J, partials);
  ncc_finalize_kernel<<<1, 256, 0, stream>>>(partials, NBLOCKS, out);

  (void)in_sizes; (void)n_in; (void)out_size; (void)ws_size;
}